// DetectionModule_28750511079888
// MI455X (gfx1250) — compile-verified
//
#include <hip/hip_runtime.h>
#include <hip/hip_bf16.h>

#define THREADS 256
#define NSUB 4          // sub-histogram copies to reduce ds_add same-address conflicts
#define MAX_BLOCKS 4096

// Native clang vector type (usable with __builtin_nontemporal_store, unlike HIP float4)
typedef float vfloat4 __attribute__((ext_vector_type(4)));

// ---------- helpers ----------

// Order-preserving float->uint map: descending float order == descending uint order.
__device__ __forceinline__ unsigned float_to_ord(float f) {
    unsigned b = __float_as_uint(f);
    return b ^ ((b & 0x80000000u) ? 0xFFFFFFFFu : 0x80000000u);
}

// ---------- workspace zeroing ----------
__global__ void init_kernel(unsigned* __restrict__ ws, int n) {
    int i = blockIdx.x * blockDim.x + threadIdx.x;
    if (i < n) ws[i] = 0u;
}

// ---------- histogram pass (levels 1..3) with async global->LDS staging ----------
// LEVEL 1: hist[u>>21]                      (2048 buckets, all elements)
// LEVEL 2: hist[(u>>10)&0x7FF]  if u>>21 == ctrl[0]            (2048 buckets)
// LEVEL 3: hist[u&0x3FF]        if u>>10 == (ctrl[0]<<11)|ctrl[2] (1024 buckets)
template <int LEVEL>
__global__ void hist_kernel(const float* __restrict__ x, long long nvec,
                            const unsigned* __restrict__ ctrl,
                            unsigned* __restrict__ ghist) {
    constexpr int NB = (LEVEL == 3) ? 1024 : 2048;
    __shared__ unsigned lh[NB * NSUB];
    __shared__ vfloat4 stage[2][THREADS];

    const int tid = threadIdx.x;
    for (int i = tid; i < NB * NSUB; i += THREADS) lh[i] = 0u;
    __syncthreads();

    unsigned match1 = 0, match2 = 0;
    if (LEVEL >= 2) match1 = ctrl[0];
    if (LEVEL == 3) match2 = (match1 << 11) | ctrl[2];

    unsigned* h = &lh[((tid >> 6) & (NSUB - 1)) * NB];

    long long i = (long long)blockIdx.x * THREADS + tid;
    const long long stride = (long long)gridDim.x * THREADS;

    // LDS byte addresses of this lane's two 16B staging slots (addrspace(3) -> offset)
    unsigned lds0 = (unsigned)(size_t)(&stage[0][tid]);
    unsigned lds1 = (unsigned)(size_t)(&stage[1][tid]);

    int parity = 0;
    if (i < nvec) {
        unsigned long long g = (unsigned long long)x + (unsigned long long)i * 16ull;
        asm volatile("global_load_async_to_lds_b128 %0, %1, off"
                     :: "v"(lds0), "v"(g) : "memory");
    }
    for (; i < nvec; i += stride) {
        long long nxt = i + stride;
        if (nxt < nvec) {
            unsigned nl = parity ? lds0 : lds1;
            unsigned long long g = (unsigned long long)x + (unsigned long long)nxt * 16ull;
            asm volatile("global_load_async_to_lds_b128 %0, %1, off"
                         :: "v"(nl), "v"(g) : "memory");
            asm volatile("s_wait_asynccnt 0x1" ::: "memory");  // current tile landed
        } else {
            asm volatile("s_wait_asynccnt 0x0" ::: "memory");
        }
        vfloat4 v = parity ? stage[1][tid] : stage[0][tid];
        parity ^= 1;

        #pragma unroll
        for (int c = 0; c < 4; ++c) {
            unsigned u = float_to_ord(v[c]);
            if (LEVEL == 1) {
                atomicAdd(&h[u >> 21], 1u);
            } else if (LEVEL == 2) {
                if ((u >> 21) == match1) atomicAdd(&h[(u >> 10) & 0x7FFu], 1u);
            } else {
                if ((u >> 10) == match2) atomicAdd(&h[u & 0x3FFu], 1u);
            }
        }
    }
    __syncthreads();

    for (int b = tid; b < NB; b += THREADS) {
        unsigned s = 0;
        #pragma unroll
        for (int c = 0; c < NSUB; ++c) s += lh[c * NB + b];
        if (s) atomicAdd(&ghist[b], s);
    }
}

// ---------- bucket scan: find bucket containing rank k (descending) ----------
// ctrl layout: [0]=b1 [1]=r2 [2]=b2 [3]=r3 [4]=thresh_bits [5]=k_ge_n flag
__global__ void scan_kernel(const unsigned* __restrict__ hist, int nb,
                            const int* __restrict__ kin,
                            unsigned* __restrict__ ctrl,
                            int level, long long n_total) {
    __shared__ unsigned sh[2048];
    __shared__ unsigned long long csum[THREADS];
    const int tid = threadIdx.x;
    const int chunk = nb / THREADS;   // 8 or 4

    unsigned long long s = 0;
    for (int j = 0; j < chunk; ++j) {
        unsigned v = hist[tid * chunk + j];
        sh[tid * chunk + j] = v;
        s += v;
    }
    csum[tid] = s;
    __syncthreads();

    if (tid != 0) return;

    long long k;
    unsigned flag;
    if (level == 1) {
        k = (long long)(*kin);
        flag = (k >= n_total) ? 1u : 0u;
        ctrl[5] = flag;
    } else {
        flag = ctrl[5];
        k = (long long)ctrl[(level == 2) ? 1 : 3];
    }
    if (flag) {  // max_features >= n: threshold is exactly 0.0
        if (level == 1)      { ctrl[0] = 0u; ctrl[1] = 0u; }
        else if (level == 2) { ctrl[2] = 0u; ctrl[3] = 0u; }
        else                 { ctrl[4] = 0u; }
        return;
    }

    long long cum = 0;
    int c = THREADS - 1;
    for (; c > 0; --c) {
        if (cum + (long long)csum[c] > k) break;
        cum += csum[c];
    }
    int b = c * chunk + chunk - 1;
    for (; b > c * chunk; --b) {
        unsigned v = sh[b];
        if (cum + (long long)v > k) break;
        cum += v;
    }
    long long rem = k - cum;

    if (level == 1)      { ctrl[0] = (unsigned)b; ctrl[1] = (unsigned)rem; }
    else if (level == 2) { ctrl[2] = (unsigned)b; ctrl[3] = (unsigned)rem; }
    else {
        unsigned u = (ctrl[0] << 21) | (ctrl[2] << 10) | (unsigned)b;
        unsigned bits = (u & 0x80000000u) ? (u ^ 0x80000000u) : ~u;
        ctrl[4] = bits;
    }
}

// ---------- apply threshold: out = (x < t) ? 0 : x ----------
__global__ void apply_kernel(const float* __restrict__ x, float* __restrict__ y,
                             long long nvec, const unsigned* __restrict__ ctrl) {
    const float t = __uint_as_float(ctrl[4]);
    long long i = (long long)blockIdx.x * THREADS + threadIdx.x;
    const long long stride = (long long)gridDim.x * THREADS;
    const vfloat4* __restrict__ xv = (const vfloat4*)x;
    vfloat4* __restrict__ yv = (vfloat4*)y;
    for (; i < nvec; i += stride) {
        if (i + stride < nvec) __builtin_prefetch(&xv[i + stride], 0, 0);
        vfloat4 v = xv[i];
        vfloat4 r;
        r.x = (v.x < t) ? 0.0f : v.x;
        r.y = (v.y < t) ? 0.0f : v.y;
        r.z = (v.z < t) ? 0.0f : v.z;
        r.w = (v.w < t) ? 0.0f : v.w;
        __builtin_nontemporal_store(r, &yv[i]);  // output never re-read: keep L2 for inputs
    }
}

// ---------- launcher ----------
static inline int nblocks(long long nvec) {
    long long b = (nvec + THREADS - 1) / THREADS;
    return (int)(b > MAX_BLOCKS ? MAX_BLOCKS : (b < 1 ? 1 : b));
}

extern "C" void kernel_launch(void* const* d_in, const int* in_sizes, int n_in,
                              void* d_out, int out_size, void* d_ws, size_t ws_size,
                              hipStream_t stream) {
    const float* early  = (const float*)d_in[0];
    const float* middle = (const float*)d_in[1];
    const float* deep   = (const float*)d_in[2];
    const int*   kmax   = (const int*)d_in[3];

    const long long n1 = in_sizes[0], n2 = in_sizes[1], n3 = in_sizes[2];
    const long long ntot = n1 + n2 + n3;
    const long long v1 = n1 / 4, v2 = n2 / 4, v3 = n3 / 4;

    float* out = (float*)d_out;
    unsigned* ws    = (unsigned*)d_ws;
    unsigned* hist1 = ws;           // 2048
    unsigned* hist2 = ws + 2048;    // 2048
    unsigned* hist3 = ws + 4096;    // 1024
    unsigned* ctrl  = ws + 5120;    // 8
    const int WSN = 5128;

    init_kernel<<<(WSN + THREADS - 1) / THREADS, THREADS, 0, stream>>>(ws, WSN);

    // Pass 1: 11-bit MSB histogram over all three tensors
    hist_kernel<1><<<nblocks(v1), THREADS, 0, stream>>>(early,  v1, ctrl, hist1);
    hist_kernel<1><<<nblocks(v2), THREADS, 0, stream>>>(middle, v2, ctrl, hist1);
    hist_kernel<1><<<nblocks(v3), THREADS, 0, stream>>>(deep,   v3, ctrl, hist1);
    scan_kernel<<<1, THREADS, 0, stream>>>(hist1, 2048, kmax, ctrl, 1, ntot);

    // Pass 2: next 11 bits, filtered to bucket b1 (inputs now L2-resident)
    hist_kernel<2><<<nblocks(v1), THREADS, 0, stream>>>(early,  v1, ctrl, hist2);
    hist_kernel<2><<<nblocks(v2), THREADS, 0, stream>>>(middle, v2, ctrl, hist2);
    hist_kernel<2><<<nblocks(v3), THREADS, 0, stream>>>(deep,   v3, ctrl, hist2);
    scan_kernel<<<1, THREADS, 0, stream>>>(hist2, 2048, kmax, ctrl, 2, ntot);

    // Pass 3: low 10 bits, filtered to (b1,b2) -> exact threshold value
    hist_kernel<3><<<nblocks(v1), THREADS, 0, stream>>>(early,  v1, ctrl, hist3);
    hist_kernel<3><<<nblocks(v2), THREADS, 0, stream>>>(middle, v2, ctrl, hist3);
    hist_kernel<3><<<nblocks(v3), THREADS, 0, stream>>>(deep,   v3, ctrl, hist3);
    scan_kernel<<<1, THREADS, 0, stream>>>(hist3, 1024, kmax, ctrl, 3, ntot);

    // Pass 4: masked copy
    apply_kernel<<<nblocks(v1), THREADS, 0, stream>>>(early,  out,           v1, ctrl);
    apply_kernel<<<nblocks(v2), THREADS, 0, stream>>>(middle, out + n1,      v2, ctrl);
    apply_kernel<<<nblocks(v3), THREADS, 0, stream>>>(deep,   out + n1 + n2, v3, ctrl);
}